// AdaConRetinaNet_28862180229655
// MI455X (gfx1250) — compile-verified
//
#include <hip/hip_runtime.h>
#include <hip/hip_bf16.h>
#include <stdint.h>

// ---------------- problem constants ----------------
#define NUM_CLASSES 80
#define B_IMG 4
#define NUM_LEVELS 5
#define TOPK_K 1000
#define NCAND 5000        // 5 levels * 1000
#define NPAD 5024         // NCAND padded to 16*314 = 32*157
#define NWORDS 157        // NPAD / 32 suppression-mask words per row
#define NIB 314           // NPAD / 16 row blocks
#define TASKS (NIB * NWORDS)
#define NSORT 8192        // next pow2 >= NCAND
#define DETS 300
#define IMG_W_F 1216.0f
#define IMG_H_F 800.0f
#define SCORE_T 0.05f
#define NMS_T 0.5f
#define NBINS 1024
#define CAND_CAP 4096
// logit(0.05) = ln(0.05/0.95)
#define LOGIT_T (-2.9444389791664403f)
#define LOGIT_HI 16.0f
#define BIN_SCALE ((float)NBINS / (LOGIT_HI - (LOGIT_T)))
// log(1000/16)
#define BCLIP 4.135166556742356f

__constant__ int c_ls[NUM_LEVELS] = {136800, 34200, 8550, 2223, 630};

struct KParams {
    const float* logit[NUM_LEVELS];
    const float* reg[NUM_LEVELS];
    const float* anc[NUM_LEVELS];
};

// ---------------- CDNA5 feature guards ----------------
#if defined(__HIP_DEVICE_COMPILE__) && __has_builtin(__builtin_amdgcn_global_load_async_to_lds_b64)
#define HAVE_ASYNC_LDS 1
#else
#define HAVE_ASYNC_LDS 0
#endif

#if defined(__HIP_DEVICE_COMPILE__) && __has_builtin(__builtin_amdgcn_tensor_load_to_lds)
#define HAVE_TDM 1
#else
#define HAVE_TDM 0
#endif

#if defined(__HIP_DEVICE_COMPILE__) && __has_builtin(__builtin_amdgcn_wmma_f32_16x16x4_f32)
#define HAVE_WMMA 1
#else
#define HAVE_WMMA 0
#endif

typedef int v2i_t __attribute__((vector_size(8)));
typedef float v2f __attribute__((ext_vector_type(2)));
typedef float v8f __attribute__((ext_vector_type(8)));

__device__ __forceinline__ void dev_wait_async() {
#if defined(__HIP_DEVICE_COMPILE__)
#if __has_builtin(__builtin_amdgcn_s_wait_asynccnt)
    __builtin_amdgcn_s_wait_asynccnt(0);
#elif HAVE_ASYNC_LDS
    asm volatile("s_wait_asynccnt 0" ::: "memory");
#endif
#endif
}

__device__ __forceinline__ void async_copy_b64(const void* g, void* l) {
#if HAVE_ASYNC_LDS
    __builtin_amdgcn_global_load_async_to_lds_b64(
        (__attribute__((address_space(1))) v2i_t*)(g),
        (__attribute__((address_space(3))) v2i_t*)(l), 0, 0);
#else
    *(unsigned long long*)l = *(const unsigned long long*)g;
#endif
}

__device__ __forceinline__ unsigned ballot32(bool p) {
#if defined(__HIP_DEVICE_COMPILE__) && __has_builtin(__builtin_amdgcn_ballot_w32)
    return __builtin_amdgcn_ballot_w32(p);
#else
    return (unsigned)__ballot(p);
#endif
}

// monotone float -> u32 key (ascending)
__device__ __forceinline__ unsigned sortable_f32(float f) {
    unsigned b = __float_as_uint(f);
    return (b & 0x80000000u) ? ~b : (b | 0x80000000u);
}
__device__ __forceinline__ float unsortable_f32(unsigned s) {
    unsigned b = (s & 0x80000000u) ? (s & 0x7FFFFFFFu) : ~s;
    return __uint_as_float(b);
}

// ascending bitonic sort of N u64 keys in LDS, N power of two
template <int N>
__device__ void bitonic_sort_lds(unsigned long long* k) {
    for (int size = 2; size <= N; size <<= 1) {
        for (int stride = size >> 1; stride > 0; stride >>= 1) {
            __syncthreads();
            for (int i = threadIdx.x; i < N; i += blockDim.x) {
                int j = i ^ stride;
                if (j > i) {
                    bool up = ((i & size) == 0);
                    unsigned long long a = k[i], b = k[j];
                    if ((a > b) == up) { k[i] = b; k[j] = a; }
                }
            }
        }
    }
    __syncthreads();
}

// ---------------- workspace layout (bytes) ----------------
static constexpr size_t HIST_OFF = 0;
static constexpr size_t HIST_BYTES = (size_t)B_IMG * NUM_LEVELS * NBINS * 4;  // 81920
static constexpr size_t CUT_OFF_B = HIST_BYTES;
static constexpr size_t CNT_OFF_B = CUT_OFF_B + 128;
static constexpr size_t ZERO_BYTES = CNT_OFF_B + 128;  // 82176, zeroed each launch
static constexpr size_t CAND_OFF_B = ZERO_BYTES;
static constexpr size_t CB_OFF_B = CAND_OFF_B + (size_t)B_IMG * NUM_LEVELS * CAND_CAP * 8;
static constexpr size_t CS_OFF_B = CB_OFF_B + (size_t)B_IMG * NCAND * 4 * 4;
static constexpr size_t CL_OFF_B = CS_OFF_B + (size_t)B_IMG * NCAND * 4;
static constexpr size_t SB_OFF_B = CL_OFF_B + (size_t)B_IMG * NCAND * 4;
static constexpr size_t SS_OFF_B = SB_OFF_B + (size_t)B_IMG * NPAD * 4 * 4;
static constexpr size_t SL_OFF_B = SS_OFF_B + (size_t)B_IMG * NPAD * 4;
static constexpr size_t SA_OFF_B = SL_OFF_B + (size_t)B_IMG * NPAD * 4;
static constexpr size_t KEEP_OFF_B = SA_OFF_B + (size_t)B_IMG * NPAD * 4;
static constexpr size_t MASK_OFF_B = KEEP_OFF_B + (size_t)B_IMG * NPAD * 4;
// MASK size = B_IMG * NPAD * NWORDS * 4  (~12.6 MB)

// ============ K1: per-(image,level) logit histogram (streaming pass 1) ============
__global__ __launch_bounds__(256) void hist_kernel(KParams P, unsigned* hist) {
    int pair = blockIdx.y;
    int b = pair / NUM_LEVELS, lvl = pair % NUM_LEVELS;
    int n = c_ls[lvl] * NUM_CLASSES;
    const float* src = P.logit[lvl] + (size_t)b * n;
    __shared__ unsigned h[NBINS];
    for (int i = threadIdx.x; i < NBINS; i += blockDim.x) h[i] = 0;
    __syncthreads();
    int step = gridDim.x * blockDim.x;
    for (int i = blockIdx.x * blockDim.x + threadIdx.x; i < n; i += step) {
        __builtin_prefetch(&src[i + step], 0, 0);  // -> global_prefetch_b8
        float x = src[i];
        if (x > LOGIT_T) {
            int bin = (int)((x - LOGIT_T) * BIN_SCALE);
            if (bin >= NBINS) bin = NBINS - 1;
            atomicAdd(&h[bin], 1u);
        }
    }
    __syncthreads();
    for (int i = threadIdx.x; i < NBINS; i += blockDim.x)
        if (h[i]) atomicAdd(&hist[(size_t)pair * NBINS + i], h[i]);
}

// ============ K2: find per-pair cutoff bin so that >= TOPK survive ============
__global__ void cutoff_kernel(const unsigned* hist, int* cutoff) {
    if (threadIdx.x == 0) {
        int pair = blockIdx.x;
        unsigned cum = 0;
        int c = 0;
        for (int bin = NBINS - 1; bin >= 0; --bin) {
            cum += hist[(size_t)pair * NBINS + bin];
            if (cum >= TOPK_K) { c = bin; break; }
        }
        cutoff[pair] = c;
    }
}

// ============ K3: compact survivors into u64 sort keys (streaming pass 2) ============
__global__ __launch_bounds__(256) void compact_kernel(KParams P, const int* cutoff,
                                                      int* cnt, unsigned long long* cand) {
    int pair = blockIdx.y;
    int b = pair / NUM_LEVELS, lvl = pair % NUM_LEVELS;
    int n = c_ls[lvl] * NUM_CLASSES;
    const float* src = P.logit[lvl] + (size_t)b * n;
    int cut = cutoff[pair];
    unsigned long long* dst = cand + (size_t)pair * CAND_CAP;
    int step = gridDim.x * blockDim.x;
    for (int i = blockIdx.x * blockDim.x + threadIdx.x; i < n; i += step) {
        float x = src[i];
        if (x > LOGIT_T) {
            int bin = (int)((x - LOGIT_T) * BIN_SCALE);
            if (bin >= NBINS) bin = NBINS - 1;
            if (bin >= cut) {
                int pos = atomicAdd(&cnt[pair], 1);
                if (pos < CAND_CAP) {
                    // ascending sort => logit desc, idx asc (matches stable top_k)
                    unsigned long long key =
                        ((unsigned long long)(~sortable_f32(x)) << 32) | (unsigned)i;
                    dst[pos] = key;
                }
            }
        }
    }
}

// ============ K4: per-pair sort (async->LDS) + take top-1000 + decode boxes ============
__global__ __launch_bounds__(1024) void sort_decode_kernel(KParams P,
                                                           const unsigned long long* cand,
                                                           const int* cnt_g, float* cBox,
                                                           float* cScore, int* cLabel) {
    int pair = blockIdx.x;
    int b = pair / NUM_LEVELS, lvl = pair % NUM_LEVELS;
    __shared__ unsigned long long key[CAND_CAP];
    int cnt = cnt_g[pair];
    if (cnt > CAND_CAP) cnt = CAND_CAP;
    const unsigned long long* src = cand + (size_t)pair * CAND_CAP;
    for (int i = threadIdx.x; i < CAND_CAP; i += blockDim.x) {
        if (i < cnt)
            async_copy_b64(&src[i], &key[i]);  // global_load_async_to_lds_b64
        else
            key[i] = ~0ull;
    }
    dev_wait_async();  // s_wait_asynccnt 0
    __syncthreads();
    bitonic_sort_lds<CAND_CAP>(key);

    for (int r = threadIdx.x; r < TOPK_K; r += blockDim.x) {
        size_t o = (size_t)b * NCAND + (size_t)lvl * TOPK_K + r;
        if (r < cnt) {
            unsigned long long kk = key[r];
            float lg = unsortable_f32(~(unsigned)(kk >> 32));
            unsigned idx = (unsigned)kk;
            unsigned anchor = idx / NUM_CLASSES;
            int label = (int)(idx - anchor * NUM_CLASSES);
            const float* rg = P.reg[lvl] + ((size_t)b * c_ls[lvl] + anchor) * 4;
            const float* an = P.anc[lvl] + (size_t)anchor * 4;
            float a0 = an[0], a1 = an[1], a2 = an[2], a3 = an[3];
            float w = a2 - a0, h = a3 - a1;
            float cx = a0 + 0.5f * w, cy = a1 + 0.5f * h;
            float dx = rg[0], dy = rg[1];
            float dw = fminf(rg[2], BCLIP), dh = fminf(rg[3], BCLIP);
            float pcx = dx * w + cx, pcy = dy * h + cy;
            float pw = expf(dw) * w, ph = expf(dh) * h;
            float x1 = fminf(fmaxf(pcx - 0.5f * pw, 0.f), IMG_W_F);
            float y1 = fminf(fmaxf(pcy - 0.5f * ph, 0.f), IMG_H_F);
            float x2 = fminf(fmaxf(pcx + 0.5f * pw, 0.f), IMG_W_F);
            float y2 = fminf(fmaxf(pcy + 0.5f * ph, 0.f), IMG_H_F);
            cBox[o * 4 + 0] = x1; cBox[o * 4 + 1] = y1;
            cBox[o * 4 + 2] = x2; cBox[o * 4 + 3] = y2;
            cScore[o] = 1.0f / (1.0f + expf(-lg));
            cLabel[o] = label;
        } else {
            cBox[o * 4 + 0] = 0.f; cBox[o * 4 + 1] = 0.f;
            cBox[o * 4 + 2] = 0.f; cBox[o * 4 + 3] = 0.f;
            cScore[o] = -1.0f;
            cLabel[o] = 0;
        }
    }
}

// ============ K5: per-image global sort + build padded arrays (box, area, label) ============
__global__ __launch_bounds__(1024) void img_sort_kernel(const float* cBox, const float* cScore,
                                                        const int* cLabel, float* sBox,
                                                        float* sScore, int* sLabel,
                                                        float* sArea) {
    int b = blockIdx.x;
    __shared__ unsigned long long key[NSORT];  // 64 KB
    for (int i = threadIdx.x; i < NSORT; i += blockDim.x) {
        key[i] = (i < NCAND)
                     ? ((unsigned long long)(~sortable_f32(cScore[(size_t)b * NCAND + i])) << 32) |
                           (unsigned)i
                     : ~0ull;
    }
    __syncthreads();
    bitonic_sort_lds<NSORT>(key);
    for (int r = threadIdx.x; r < NPAD; r += blockDim.x) {
        size_t so = (size_t)b * NPAD + r;
        if (r < NCAND) {
            int j = (int)(unsigned)key[r];
            size_t co = (size_t)b * NCAND + j;
            float x1 = cBox[co * 4 + 0], y1 = cBox[co * 4 + 1];
            float x2 = cBox[co * 4 + 2], y2 = cBox[co * 4 + 3];
            sScore[so] = cScore[co];
            sLabel[so] = cLabel[co];
            sBox[so * 4 + 0] = x1;
            sBox[so * 4 + 1] = y1;
            sBox[so * 4 + 2] = x2;
            sBox[so * 4 + 3] = y2;
            sArea[so] = (x2 - x1) * (y2 - y1);
        } else {  // padding rows so mask tiles are full
            sScore[so] = -1.0f;
            sLabel[so] = -1;
            sBox[so * 4 + 0] = 0.f; sBox[so * 4 + 1] = 0.f;
            sBox[so * 4 + 2] = 0.f; sBox[so * 4 + 3] = 0.f;
            sArea[so] = 0.f;
        }
    }
}

// ============ K6a: WMMA-assisted suppression bitmask matrix ============
// Each wave handles one 16(i) x 32(j) tile. area_i + area_j for a 16x16 tile is a
// rank-2 product computed by V_WMMA_F32_16X16X4_F32:
//   A(16x4) = [area_i, 1, 0, 0],  B(4x16) = [1; area_j; 0; 0]  =>  D[i][j] = a_i + a_j
// VALU computes the min/max intersection, ballot packs 32 suppression bits per row word.
__global__ __launch_bounds__(256) void mask_kernel(const float* sBox, const float* sArea,
                                                   const int* sLabel, unsigned* mask) {
    int b = blockIdx.y;
    const float4* pB4 = (const float4*)(sBox) + (size_t)b * NPAD;
    const float* pA = sArea + (size_t)b * NPAD;
    const int* pL = sLabel + (size_t)b * NPAD;
    unsigned* pM = mask + (size_t)b * NPAD * NWORDS;

    int lane = threadIdx.x & 31;
    int wid = threadIdx.x >> 5;
    int wavesPerBlock = blockDim.x >> 5;
    int waveStride = gridDim.x * wavesPerBlock;
    int jrel = lane & 15;
    bool lo = lane < 16;

    for (int task = blockIdx.x * wavesPerBlock + wid; task < TASKS; task += waveStride) {
        int ib = task / NWORDS;
        int jw = task - ib * NWORDS;
        int i0 = ib * 16;
        int j0 = jw * 32;

        // ---- area sums for both 16x16 tiles ----
        float aI = pA[i0 + jrel];
        float aJ0 = pA[j0 + jrel];
        float aJ1 = pA[j0 + 16 + jrel];
#if HAVE_WMMA
        v2f A;  A.x = lo ? aI : 0.f;   A.y = lo ? 1.0f : 0.f;   // K0/K1 (K2/K3 = 0)
        v2f B0; B0.x = lo ? 1.0f : 0.f; B0.y = lo ? aJ0 : 0.f;
        v2f B1; B1.x = lo ? 1.0f : 0.f; B1.y = lo ? aJ1 : 0.f;
        v8f cz = {0.f, 0.f, 0.f, 0.f, 0.f, 0.f, 0.f, 0.f};
        v8f d0 = __builtin_amdgcn_wmma_f32_16x16x4_f32(false, A, false, B0, (short)0, cz,
                                                       false, false);
        v8f d1 = __builtin_amdgcn_wmma_f32_16x16x4_f32(false, A, false, B1, (short)0, cz,
                                                       false, false);
#endif
        int jA = j0 + jrel;
        int jB = jA + 16;
        float4 bj0 = pB4[jA];
        float4 bj1 = pB4[jB];
        int lj0 = pL[jA];
        int lj1 = pL[jB];

#pragma unroll
        for (int v = 0; v < 8; ++v) {
            int i = i0 + v + (lo ? 0 : 8);
            float4 bi = pB4[i];
            int li = pL[i];
            float s0, s1;
#if HAVE_WMMA
            s0 = d0[v];
            s1 = d1[v];
#else
            float ai = pA[i];
            s0 = ai + pA[jA];
            s1 = ai + pA[jB];
#endif
            // tile A (columns j0 .. j0+15)
            float iw = fmaxf(fminf(bi.z, bj0.z) - fmaxf(bi.x, bj0.x), 0.f);
            float ih = fmaxf(fminf(bi.w, bj0.w) - fmaxf(bi.y, bj0.y), 0.f);
            float inter = iw * ih;
            float iou = inter / (s0 - inter);
            bool pAp = (jA > i) && (lj0 == li) && (iou > NMS_T);
            // tile B (columns j0+16 .. j0+31)
            iw = fmaxf(fminf(bi.z, bj1.z) - fmaxf(bi.x, bj1.x), 0.f);
            ih = fmaxf(fminf(bi.w, bj1.w) - fmaxf(bi.y, bj1.y), 0.f);
            inter = iw * ih;
            iou = inter / (s1 - inter);
            bool pBp = (jB > i) && (lj1 == li) && (iou > NMS_T);

            unsigned mA = ballot32(pAp);   // [15:0]=row v bits, [31:16]=row v+8 bits
            unsigned mB = ballot32(pBp);
            if (lane == 0) {
                pM[(size_t)(i0 + v) * NWORDS + jw] = (mA & 0xFFFFu) | (mB << 16);
                pM[(size_t)(i0 + v + 8) * NWORDS + jw] = (mA >> 16) | (mB & 0xFFFF0000u);
            }
        }
    }
}

// ============ K6b: sequential bitset reduction (scores staged via TDM) ============
__global__ __launch_bounds__(256) void nms_reduce_kernel(const float* sScore,
                                                         const unsigned* mask, int* keepg) {
    int b = blockIdx.x;
    __shared__ float scoreLds[NPAD];     // ~20 KB, DMA'd by the Tensor Data Mover
    __shared__ unsigned removed[NWORDS];

#if HAVE_TDM
    {
        typedef unsigned int v4u __attribute__((ext_vector_type(4)));
        typedef int v4i __attribute__((ext_vector_type(4)));
        typedef int v8i __attribute__((ext_vector_type(8)));
        if (threadIdx.x < 32) {  // one wave issues the TDM DMA (EXEC ignored by TDM)
            unsigned long long ga =
                (unsigned long long)(size_t)(const void*)(sScore + (size_t)b * NPAD);
            unsigned la = (unsigned)(size_t)(__attribute__((address_space(3))) float*)scoreLds;
            v4u g0;
            g0.x = 1u;  // count=1, user descriptor
            g0.y = la;  // lds_addr
            g0.z = (unsigned)ga;
            g0.w = (unsigned)((ga >> 32) & 0x01FFFFFFu) | 0x80000000u;  // type=2 (image)
            v8i g1;
            g1[0] = 0x00020000;         // data_size=4B, no multicast
            g1[1] = (int)(NPAD << 16);  // tensor_dim0 low16
            g1[2] = (1 << 16);          // tensor_dim0 hi16=0 | tensor_dim1=1
            g1[3] = (int)(NPAD << 16);  // tile_dim0
            g1[4] = 0;                  // tile_dim1/2 unused (1D)
            g1[5] = NPAD;               // tensor_dim0_stride low32
            g1[6] = 0;
            g1[7] = 0;
            v4i z4 = {0, 0, 0, 0};
#if __clang_major__ >= 23
            v8i z8 = {0, 0, 0, 0, 0, 0, 0, 0};
            __builtin_amdgcn_tensor_load_to_lds(g0, g1, z4, z4, z8, 0);
#else
            __builtin_amdgcn_tensor_load_to_lds(g0, g1, z4, z4, 0);
#endif
#if __has_builtin(__builtin_amdgcn_s_wait_tensorcnt)
            __builtin_amdgcn_s_wait_tensorcnt(0);
#else
            asm volatile("s_wait_tensorcnt 0" ::: "memory");
#endif
        }
    }
#else
    for (int i = threadIdx.x; i < NPAD; i += blockDim.x)
        scoreLds[i] = sScore[(size_t)b * NPAD + i];
#endif
    for (int t = threadIdx.x; t < NWORDS; t += blockDim.x) removed[t] = 0;
    __syncthreads();

    const unsigned* pM = mask + (size_t)b * NPAD * NWORDS;
    for (int i = 0; i < NCAND; ++i) {
        __syncthreads();  // make previous iteration's ORs visible
        unsigned w = removed[i >> 5];
        bool sup = (w >> (i & 31)) & 1u;
        bool kept = !sup && (scoreLds[i] > SCORE_T);  // uniform across block
        if (threadIdx.x == 0) keepg[(size_t)b * NPAD + i] = kept ? 1 : 0;
        if (kept) {
            const unsigned* row = pM + (size_t)i * NWORDS;
            for (int t = threadIdx.x; t < NWORDS; t += blockDim.x) removed[t] |= row[t];
        }
    }
}

// ============ K7: emit top-300 detections per image ============
__global__ __launch_bounds__(512) void final_kernel(const float* sBox, const float* sScore,
                                                    const int* sLabel, const int* keep,
                                                    float* out) {
    int b = blockIdx.x;
    __shared__ int sel[DETS];
    __shared__ int flag[DETS];
    if (threadIdx.x == 0) {
        int c = 0;
        for (int j = 0; j < NCAND && c < DETS; ++j)
            if (keep[(size_t)b * NPAD + j]) { sel[c] = j; flag[c] = 1; ++c; }
        for (int j = 0; j < NCAND && c < DETS; ++j)
            if (!keep[(size_t)b * NPAD + j]) { sel[c] = j; flag[c] = 0; ++c; }
    }
    __syncthreads();
    for (int r = threadIdx.x; r < DETS; r += blockDim.x) {
        int j = sel[r];
        int f = flag[r];
        size_t so = (size_t)b * NPAD + j;
        out[(size_t)b * DETS * 4 + r * 4 + 0] = sBox[so * 4 + 0];
        out[(size_t)b * DETS * 4 + r * 4 + 1] = sBox[so * 4 + 1];
        out[(size_t)b * DETS * 4 + r * 4 + 2] = sBox[so * 4 + 2];
        out[(size_t)b * DETS * 4 + r * 4 + 3] = sBox[so * 4 + 3];
        out[(size_t)B_IMG * DETS * 4 + (size_t)b * DETS + r] = f ? sScore[so] : -1.0f;
        ((int*)out)[(size_t)B_IMG * DETS * 5 + (size_t)b * DETS + r] = sLabel[so];
    }
}

// ---------------- host side ----------------
extern "C" void kernel_launch(void* const* d_in, const int* in_sizes, int n_in, void* d_out,
                              int out_size, void* d_ws, size_t ws_size, hipStream_t stream) {
    (void)in_sizes; (void)n_in; (void)out_size; (void)ws_size;

    KParams P;
    for (int i = 0; i < NUM_LEVELS; ++i) {
        P.logit[i] = (const float*)d_in[3 * i + 0];
        P.reg[i]   = (const float*)d_in[3 * i + 1];
        P.anc[i]   = (const float*)d_in[3 * i + 2];
    }

    char* ws = (char*)d_ws;
    unsigned* hist           = (unsigned*)(ws + HIST_OFF);
    int* cutoff              = (int*)(ws + CUT_OFF_B);
    int* cand_cnt            = (int*)(ws + CNT_OFF_B);
    unsigned long long* cand = (unsigned long long*)(ws + CAND_OFF_B);
    float* cBox              = (float*)(ws + CB_OFF_B);
    float* cScore            = (float*)(ws + CS_OFF_B);
    int* cLabel              = (int*)(ws + CL_OFF_B);
    float* sBox              = (float*)(ws + SB_OFF_B);
    float* sScore            = (float*)(ws + SS_OFF_B);
    int* sLabel              = (int*)(ws + SL_OFF_B);
    float* sArea             = (float*)(ws + SA_OFF_B);
    int* keepg               = (int*)(ws + KEEP_OFF_B);
    unsigned* mask           = (unsigned*)(ws + MASK_OFF_B);
    float* out               = (float*)d_out;

    (void)hipMemsetAsync(d_ws, 0, ZERO_BYTES, stream);

    dim3 scanGrid(1024, B_IMG * NUM_LEVELS, 1);
    hist_kernel<<<scanGrid, 256, 0, stream>>>(P, hist);
    cutoff_kernel<<<B_IMG * NUM_LEVELS, 64, 0, stream>>>(hist, cutoff);
    compact_kernel<<<scanGrid, 256, 0, stream>>>(P, cutoff, cand_cnt, cand);
    sort_decode_kernel<<<B_IMG * NUM_LEVELS, 1024, 0, stream>>>(P, cand, cand_cnt, cBox, cScore,
                                                                cLabel);
    img_sort_kernel<<<B_IMG, 1024, 0, stream>>>(cBox, cScore, cLabel, sBox, sScore, sLabel,
                                                sArea);
    mask_kernel<<<dim3(1024, B_IMG, 1), 256, 0, stream>>>(sBox, sArea, sLabel, mask);
    nms_reduce_kernel<<<B_IMG, 256, 0, stream>>>(sScore, mask, keepg);
    final_kernel<<<B_IMG, 512, 0, stream>>>(sBox, sScore, sLabel, keepg, out);
}